// OutOfDistributionDetector_53867479827172
// MI455X (gfx1250) — compile-verified
//
#include <hip/hip_runtime.h>
#include <hip/hip_bf16.h>

typedef __attribute__((ext_vector_type(2))) float v2f;
typedef __attribute__((ext_vector_type(8))) float v8f;

#define NCLS 10

// ---------------- deterministic RNG (eps only enters as eps^2) ----------------
__device__ __forceinline__ unsigned pcg_hash(unsigned v) {
  unsigned s = v * 747796405u + 2891336453u;
  unsigned w = ((s >> ((s >> 28u) + 4u)) ^ s) * 277803737u;
  return (w >> 22u) ^ w;
}

__device__ __forceinline__ float eps_sq(unsigned idx) {
  unsigned h1 = pcg_hash(idx * 2u + 1u);
  unsigned h2 = pcg_hash(idx * 2u + 2u);
  float u1 = ((float)h1 + 0.5f) * 2.3283064365386963e-10f;
  float u2 = ((float)h2 + 0.5f) * 2.3283064365386963e-10f;
  float r2 = -2.0f * __logf(u1);               // Box-Muller radius^2
  float c  = __cosf(6.28318530717958647692f * u2);
  return r2 * c * c;                            // eps^2
}

// ---------------- fused branch kernel: one workgroup per image ----------------
// Per branch: xi = x + noise-conv (LDS, zero-padded 34x34), then per 16ch x 16sp
// tile:  f = W x patches  via 7x V_WMMA_F32_16X16X4_F32 (implicit GEMM, K=27->28)
//        f --LDS transpose--> A-layout, then x_mean/x_std via 8x WMMA vs mu/std.
template<int BRANCH, int C, int S, int STRIDE>
__global__ void __launch_bounds__(256) ood_branch_kernel(
    const float* __restrict__ x,      const float* __restrict__ noise_w,
    const float* __restrict__ noise_b,
    const float* __restrict__ feat_w, const float* __restrict__ feat_b,
    const float* __restrict__ mu_w,   const float* __restrict__ mu_b,
    const float* __restrict__ std_w,  const float* __restrict__ std_b,
    float* __restrict__ mean_out, float* __restrict__ std_out,
    float* __restrict__ scores)
{
  constexpr int WO    = 32 / STRIDE;   // output width; S == WO*WO
  constexpr int CT    = C / 16;        // channel tiles of 16
  constexpr int PADLO = (STRIDE == 1) ? 1 : 0;   // jax SAME padding
  const int b    = blockIdx.x;
  const int tid  = threadIdx.x;
  const int lane = tid & 31;
  const int wave = tid >> 5;
  const int half = lane >> 4;          // which 16-lane group (K-pair selector)
  const int m    = lane & 15;          // A/B row-col index within 16

  extern __shared__ float smem[];
  float* x_s     = smem;                  // 3072   raw image
  float* xi_p    = x_s + 3072;            // 3*34*34 zero-padded noised image
  float* mu_t    = xi_p + 3468;           // 10*S   mu_w transposed [s][k]
  float* std_t   = mu_t + 10 * S;         // 10*S
  float* mean_sh = std_t + 10 * S;        // 10*C   (seeded with mu_b)
  float* std_sh  = mean_sh + 10 * C;      // 10*C   (seeded with std_b)
  float* pstage  = std_sh + 10 * C;       // 2048   spatial-split partials
  float* red_sh  = pstage + 2048;         // 2560   mahal reduction / ft overlay
  float* sum_sh  = red_sh + 2560;         // 10
  int*   kb_sh   = (int*)(sum_sh + 10);   // 1
  float* ft      = red_sh + wave * 272;   // wave-local 16x17 f-tile (transpose)

  const float* xb = x + (size_t)b * 3072;

  // ---- stage everything into LDS --------------------------------------------
  for (int p = tid; p < 3072; p += 256) x_s[p]  = xb[p];
  for (int p = tid; p < 3468; p += 256) xi_p[p] = 0.f;
  for (int p = tid; p < 10 * S; p += 256) {
    int k = p / S, s = p - k * S;        // coalesced global read, [s][k] write
    mu_t [s * 10 + k] = mu_w [p];
    std_t[s * 10 + k] = std_w[p];
  }
  for (int p = tid; p < 10 * C; p += 256) {
    int k = p / C;
    mean_sh[p] = mu_b[k];
    std_sh[p]  = std_b[k];
  }
  __syncthreads();

  // ---- noise conv: xi[c] = x[c] + conv(x, noise_w)[3*BRANCH+c] + noise_b ----
  for (int p = tid; p < 3072; p += 256) {
    int c = p >> 10, pos = p & 1023;
    int oy = pos >> 5, ox = pos & 31;
    const float* nw = noise_w + (3 * BRANCH + c) * 27;
    float acc = noise_b[3 * BRANCH + c] + x_s[p];
#pragma unroll
    for (int cc = 0; cc < 3; ++cc)
#pragma unroll
      for (int dy = 0; dy < 3; ++dy) {
        int iy = oy - 1 + dy;
        if (iy < 0 || iy > 31) continue;
#pragma unroll
        for (int dx = 0; dx < 3; ++dx) {
          int ix = ox - 1 + dx;
          if (ix < 0 || ix > 31) continue;
          acc = fmaf(nw[cc * 9 + dy * 3 + dx], x_s[cc * 1024 + iy * 32 + ix], acc);
        }
      }
    xi_p[c * 1156 + (oy + 1) * 34 + (ox + 1)] = acc;
  }
  __syncthreads();

  // ---- per-lane patch offsets for the conv GEMM B fragments -----------------
  // patch row k = (cc,dy,dx); this lane owns K = 4*st + 2*half + j, j=0/1.
  // k==27 (pad) -> offset 0; its A component is zeroed, so product is 0.
  int poff[14];
#pragma unroll
  for (int st = 0; st < 7; ++st)
#pragma unroll
    for (int j = 0; j < 2; ++j) {
      int k = 4 * st + 2 * half + j;
      poff[2 * st + j] =
          (k < 27) ? (k / 9) * 1156 + ((k % 9) / 3) * 34 + (k % 3) : 0;
    }

  // ---- fused conv-GEMM + contraction-GEMM per 16ch x 16sp tile --------------
  auto run_tile = [&](int c_base, int s_lo, int s_hi, v8f& acc_mu, v8f& acc_sd) {
    const int cA = c_base + m;                 // channel in A fragments
    v2f a_w[7];                                // W in A layout, K tiles of 4
#pragma unroll
    for (int st = 0; st < 7; ++st) {
      int k0 = 4 * st + 2 * half;
      a_w[st].x = (k0     < 27) ? feat_w[cA * 27 + k0    ] : 0.f;
      a_w[st].y = (k0 + 1 < 27) ? feat_w[cA * 27 + k0 + 1] : 0.f;
    }
    float fbr[8];                              // bias per D row channel
#pragma unroll
    for (int r = 0; r < 8; ++r) fbr[r] = feat_b[c_base + r + 8 * half];

    for (int sb = s_lo; sb < s_hi; sb += 16) {
      // per-lane patch base: this lane supplies B column N = m (spatial sb+m)
      const int s  = sb + m;
      const int oy = s / WO, ox = s % WO;
      const int pbase = (oy * STRIDE - PADLO + 1) * 34
                      + (ox * STRIDE - PADLO + 1);

      // implicit-GEMM conv: f[c][sb+m], K = 27 (padded to 28)
      v8f facc = {};
#pragma unroll
      for (int st = 0; st < 7; ++st) {
        v2f bp;
        bp.x = xi_p[pbase + poff[2 * st]];
        bp.y = xi_p[pbase + poff[2 * st + 1]];
        facc = __builtin_amdgcn_wmma_f32_16x16x4_f32(false, a_w[st], false, bp,
                                                     (short)0, facc, false, false);
      }

      // transpose f tile through wave-local LDS, folding in feat bias.
      // D layout: vgpr r -> channel r+8*half, lane -> spatial m.
#pragma unroll
      for (int r = 0; r < 8; ++r)
        ft[(r + 8 * half) * 17 + m] = facc[r] + fbr[r];
      // same-wave LDS ops are in-order: safe to read back immediately.

      // contraction: A = f (M=channel via lane m, K=s_local), B = mu/std frags
#pragma unroll
      for (int st = 0; st < 4; ++st) {
        const int kl0 = 4 * st + 2 * half;
        v2f af;
        af.x = ft[m * 17 + kl0];
        af.y = ft[m * 17 + kl0 + 1];
        const int s0 = sb + kl0, s1 = s0 + 1;
        v2f bm = {}, bs = {};
        if (m < NCLS) {
          bm.x = mu_t [s0 * 10 + m];  bm.y = mu_t [s1 * 10 + m];
          bs.x = std_t[s0 * 10 + m];  bs.y = std_t[s1 * 10 + m];
        }
        acc_mu = __builtin_amdgcn_wmma_f32_16x16x4_f32(false, af, false, bm,
                                                       (short)0, acc_mu, false, false);
        acc_sd = __builtin_amdgcn_wmma_f32_16x16x4_f32(false, af, false, bs,
                                                       (short)0, acc_sd, false, false);
      }
    }
  };
  // D layout of contraction: vgpr r -> channel c_base+r+8*half, column N = m
  auto commit = [&](int c_base, const v8f& am, const v8f& as) {
    if (m < NCLS) {
#pragma unroll
      for (int r = 0; r < 8; ++r) {
        int cw = c_base + r + 8 * half;
        mean_sh[m * C + cw] += am[r];   // single writer per element
        std_sh [m * C + cw] += as[r];
      }
    }
  };

  if constexpr (CT >= 8) {
    for (int t = wave; t < CT; t += 8) {
      v8f am = {}, as = {};
      run_tile(t * 16, 0, S, am, as);
      commit(t * 16, am, as);
    }
    __syncthreads();
  } else {                              // C==64: 4 tiles, split S across 2 waves
    constexpr int NSW = 8 / CT;
    constexpr int CH  = S / NSW;
    const int t   = wave % CT;
    const int cid = wave / CT;
    v8f am = {}, as = {};
    run_tile(t * 16, cid * CH, (cid + 1) * CH, am, as);
    if (cid == 1) {
#pragma unroll
      for (int r = 0; r < 8; ++r) {
        pstage[           (t * 32 + lane) * 8 + r] = am[r];
        pstage[CT * 256 + (t * 32 + lane) * 8 + r] = as[r];
      }
    }
    __syncthreads();
    if (cid == 0) {
#pragma unroll
      for (int r = 0; r < 8; ++r) {
        am[r] += pstage[           (t * 32 + lane) * 8 + r];
        as[r] += pstage[CT * 256 + (t * 32 + lane) * 8 + r];
      }
      commit(t * 16, am, as);
    }
    __syncthreads();
  }

  // ---- mahalanobis: mahal[k] = -0.5 * sum_c std^2 * eps^2 -------------------
  {
    float part[NCLS];
#pragma unroll
    for (int k = 0; k < NCLS; ++k) part[k] = 0.f;
    const unsigned base = ((unsigned)b * 5u + (unsigned)BRANCH) * (NCLS * 512u);
    for (int c = tid; c < C; c += 256) {
#pragma unroll
      for (int k = 0; k < NCLS; ++k) {
        float sd = std_sh[k * C + c];
        part[k] = fmaf(sd * sd, eps_sq(base + (unsigned)k * 512u + (unsigned)c),
                       part[k]);
      }
    }
#pragma unroll
    for (int k = 0; k < NCLS; ++k) red_sh[k * 256 + tid] = part[k];
  }
  __syncthreads();
  if (tid < NCLS) {                      // deterministic serial reduction
    float s = 0.f;
    for (int t = 0; t < 256; ++t) s += red_sh[tid * 256 + t];
    sum_sh[tid] = s;
  }
  __syncthreads();
  if (tid == 0) {                        // argmax(mahal) == argmin(sum)
    int kb = 0; float mn = sum_sh[0];
#pragma unroll
    for (int k = 1; k < NCLS; ++k)
      if (sum_sh[k] < mn) { mn = sum_sh[k]; kb = k; }
    *kb_sh = kb;
    scores[b * 5 + BRANCH] = -0.5f * mn;
  }
  __syncthreads();
  const int kb = *kb_sh;
  for (int c = tid; c < C; c += 256) {
    mean_out[(size_t)b * C + c] = mean_sh[kb * C + c];
    std_out [(size_t)b * C + c] = std_sh [kb * C + c];
  }
}

// ---------------- final tiny classifier: out = scores @ clf_w.T + clf_b ------
__global__ void ood_clf_kernel(const float* __restrict__ scores,
                               const float* __restrict__ clf_w,
                               const float* __restrict__ clf_b,
                               float* __restrict__ out) {
  int bidx = blockIdx.x * blockDim.x + threadIdx.x;
  if (bidx < 1024) {
    float acc = clf_b[0];
#pragma unroll
    for (int i = 0; i < 5; ++i) acc = fmaf(scores[bidx * 5 + i], clf_w[i], acc);
    out[bidx] = acc;
  }
}

extern "C" void kernel_launch(void* const* d_in, const int* in_sizes, int n_in,
                              void* d_out, int out_size, void* d_ws, size_t ws_size,
                              hipStream_t stream) {
  (void)in_sizes; (void)n_in; (void)out_size; (void)ws_size;
  const float* x       = (const float*)d_in[0];
  const float* noise_w = (const float*)d_in[1];
  const float* noise_b = (const float*)d_in[2];
  const float* clf_w   = (const float*)d_in[33];
  const float* clf_b   = (const float*)d_in[34];
  float* out    = (float*)d_out;
  float* scores = (float*)d_ws;          // 1024*5 floats of scratch

  const int Cs[5] = {64, 64, 128, 256, 512};
  float* mean_ptr[5]; float* std_ptr[5];
  size_t off = 1024;
  for (int i = 0; i < 5; ++i) { mean_ptr[i] = out + off; off += (size_t)1024 * Cs[i]; }
  for (int i = 0; i < 5; ++i) { std_ptr[i]  = out + off; off += (size_t)1024 * Cs[i]; }

#define LAUNCH_BRANCH(I, CC, SS, ST)                                              \
  do {                                                                            \
    size_t smem = (size_t)(3072 + 3468 + 20 * (SS) + 20 * (CC) + 2048 + 2560 + 16)\
                  * sizeof(float);                                                \
    hipFuncSetAttribute(                                                          \
        reinterpret_cast<const void*>(&ood_branch_kernel<I, CC, SS, ST>),         \
        hipFuncAttributeMaxDynamicSharedMemorySize, (int)smem);                   \
    ood_branch_kernel<I, CC, SS, ST><<<1024, 256, smem, stream>>>(                \
        x, noise_w, noise_b,                                                      \
        (const float*)d_in[3 + 6 * I], (const float*)d_in[4 + 6 * I],             \
        (const float*)d_in[5 + 6 * I], (const float*)d_in[6 + 6 * I],             \
        (const float*)d_in[7 + 6 * I], (const float*)d_in[8 + 6 * I],             \
        mean_ptr[I], std_ptr[I], scores);                                         \
  } while (0)

  LAUNCH_BRANCH(0,  64, 1024, 1);
  LAUNCH_BRANCH(1,  64, 1024, 1);
  LAUNCH_BRANCH(2, 128,  256, 2);
  LAUNCH_BRANCH(3, 256,   64, 4);
  LAUNCH_BRANCH(4, 512,   16, 8);
#undef LAUNCH_BRANCH

  ood_clf_kernel<<<4, 256, 0, stream>>>(scores, clf_w, clf_b, out);
}